// PoissonNoise_27547920236954
// MI455X (gfx1250) — compile-verified
//
#include <hip/hip_runtime.h>
#include <stdint.h>

// ---------------------------------------------------------------------------
// PoissonNoise for MI455X (gfx1250).
// Memory-bound streaming op: 402MB traffic -> ~17us @ 23.3TB/s roofline.
// Data path: global_load_async_to_lds_b128 double-buffered staging (ASYNCcnt),
// ds_load_b128 consume, non-temporal b128 stores (working set > 192MB L2).
// Sampler: PTRS transformed rejection (lam>=10), Knuth inversion (lam<10),
// counter-based PCG hash RNG keyed on element index (deterministic).
// ---------------------------------------------------------------------------

#define TPB 256

typedef int   v4i   __attribute__((vector_size(16)));     // matches builtin param
typedef float fvec4 __attribute__((ext_vector_type(4)));  // native clang vector

// ---------------- counter-based RNG -----------------------------------------
__device__ __forceinline__ uint32_t mix32(uint32_t x) {
    x ^= x >> 17; x *= 0xed5ad4bbu;
    x ^= x >> 11; x *= 0xac4c1b51u;
    x ^= x >> 15; x *= 0x31848babu;
    x ^= x >> 14;
    return x;
}

struct Rng {
    uint32_t s;
    __device__ __forceinline__ void init(uint32_t idx) { s = mix32(idx ^ 0x9e3779b9u); }
    __device__ __forceinline__ float next() {            // uniform in (0,1]
        s = s * 1664525u + 1013904223u;
        uint32_t z = mix32(s);
        return (float)((z >> 8) + 1u) * 0x1p-24f;
    }
};

// ln(k!) via Stirling with 1/(12x) correction; exact-enough for k>=2.
__device__ __forceinline__ float logfactf(float k) {
    if (k < 2.f) return 0.f;                       // 0! = 1! = 1
    float x = k + 1.f;
    return (x - 0.5f) * __logf(x) - x + 0.918938533f + (1.f / 12.f) / x;
}

// Poisson sampler: Knuth inversion for small lam, Hormann PTRS for lam>=10.
__device__ __forceinline__ float poisson_sample(float lam, Rng& r) {
    if (lam < 10.f) {
        if (lam <= 0.f) return 0.f;
        float L = __expf(-lam);
        float p = 1.f;
        float k = -1.f;
#pragma unroll 1
        for (int i = 0; i < 64; ++i) {
            k += 1.f;
            p *= r.next();
            if (p <= L) break;
        }
        return k;
    }
    // PTRS (Hormann 1993 transformed rejection)
    const float slam      = __fsqrt_rn(lam);
    const float loglam    = __logf(lam);
    const float b         = 0.931f + 2.53f * slam;
    const float a         = -0.059f + 0.02483f * b;
    const float inv_alpha = 1.1239f + 1.1328f / (b - 3.4f);
    const float vr        = 0.9277f - 3.6224f / (b - 2.f);
#pragma unroll 1
    for (int i = 0; i < 64; ++i) {
        float U  = r.next() - 0.5f;
        float V  = r.next();
        float us = 0.5f - fabsf(U);
        float k  = floorf((2.f * a / us + b) * U + lam + 0.43f);
        if (us >= 0.07f && V <= vr) return k;                 // fast accept (~90%)
        if (k < 0.f || (us < 0.013f && V > us)) continue;     // fast reject
        float lhs = __logf(V) + __logf(inv_alpha) - __logf(a / (us * us) + b);
        float rhs = k * loglam - lam - logfactf(k);
        if (lhs <= rhs) return k;
    }
    return floorf(lam + 0.5f);  // unreachable in practice
}

// map one input element -> output element
__device__ __forceinline__ float poisson_noise_elem(float d, uint32_t eidx) {
    Rng r; r.init(eidx);
    float x   = fminf(fmaxf((d + 1.f) * 0.5f, 0.f), 1.f);
    float lam = x * 255.f;
    float k   = poisson_sample(lam, r);
    float y   = k * (2.f / 255.f) - 1.f;
    return fminf(fmaxf(y, -1.f), 1.f);
}

// ---------------- async global->LDS staging ----------------------------------
__device__ __forceinline__ void async_copy_b128(const void* gsrc, void* ldst) {
    __builtin_amdgcn_global_load_async_to_lds_b128(
        (__attribute__((address_space(1))) v4i*)gsrc,
        (__attribute__((address_space(3))) v4i*)ldst,
        0 /*offset*/, 0 /*cpol*/);
}

// ---------------- main kernel ------------------------------------------------
__global__ void __launch_bounds__(TPB) poisson_noise_kernel(
    const float* __restrict__ in, float* __restrict__ out, int n4, int tiles)
{
    __shared__ fvec4 buf[2][TPB];
    const int tid    = threadIdx.x;
    const int stride = (int)gridDim.x;
    int t  = (int)blockIdx.x;
    int pb = 0;

    if (t < tiles) {
        int idx4 = t * TPB + tid;
        if (idx4 > n4 - 1) idx4 = n4 - 1;               // clamp: safe OOB guard
        async_copy_b128(in + (size_t)idx4 * 4, &buf[0][tid]);
    }

#pragma unroll 1
    for (; t < tiles; t += stride, pb ^= 1) {
        const int tn = t + stride;
        if (tn < tiles) {
            // previous consumer of buf[pb^1] must have drained its ds_load
            asm volatile("s_wait_dscnt 0" ::: "memory");
            int idx4n = tn * TPB + tid;
            if (idx4n > n4 - 1) idx4n = n4 - 1;
            async_copy_b128(in + (size_t)idx4n * 4, &buf[pb ^ 1][tid]);
            // loads retire in order: <=1 outstanding means current tile landed
            asm volatile("s_wait_asynccnt 1" ::: "memory");
        } else {
            asm volatile("s_wait_asynccnt 0" ::: "memory");
        }

        const fvec4 v  = buf[pb][tid];                   // ds_load_b128
        const int idx4 = t * TPB + tid;
        if (idx4 < n4) {
            const uint32_t e = (uint32_t)idx4 * 4u;
            fvec4 o;
            o.x = poisson_noise_elem(v.x, e + 0u);
            o.y = poisson_noise_elem(v.y, e + 1u);
            o.z = poisson_noise_elem(v.z, e + 2u);
            o.w = poisson_noise_elem(v.w, e + 3u);
            __builtin_nontemporal_store(o, (fvec4*)(out + (size_t)idx4 * 4));
        }
    }
}

// scalar tail for n % 4 != 0 (not hit for 64*3*512*512, kept for generality)
__global__ void poisson_noise_tail(const float* __restrict__ in,
                                   float* __restrict__ out, int start, int n)
{
    int i = start + (int)(blockIdx.x * blockDim.x + threadIdx.x);
    if (i < n) out[i] = poisson_noise_elem(in[i], (uint32_t)i);
}

extern "C" void kernel_launch(void* const* d_in, const int* in_sizes, int n_in,
                              void* d_out, int out_size, void* d_ws, size_t ws_size,
                              hipStream_t stream)
{
    (void)n_in; (void)d_ws; (void)ws_size; (void)out_size;
    const float* in  = (const float*)d_in[0];
    float*       out = (float*)d_out;

    const int n     = in_sizes[0];
    const int n4    = n >> 2;
    const int rem   = n - (n4 << 2);
    const int tiles = (n4 + TPB - 1) / TPB;

    int grid = tiles < 4096 ? tiles : 4096;   // persistent blocks, 8KB LDS each
    if (grid > 0)
        poisson_noise_kernel<<<grid, TPB, 0, stream>>>(in, out, n4, tiles);
    if (rem > 0)
        poisson_noise_tail<<<1, 64, 0, stream>>>(in, out, n4 << 2, n);
}